// TopKLayer_58222576664882
// MI455X (gfx1250) — compile-verified
//
#include <hip/hip_runtime.h>

// Problem constants (match the reference exactly).
#define TPB     256
#define L_ROW   8192          // elements per row
#define B_ROWS  8192          // rows
#define K_TOP   4096          // rank from the top (L * (1 - sigmoid(0)) = L/2)
#define N_SEED  128           // rows pre-selected exactly to seed the pruning threshold
#define VEC_PER_T (L_ROW / (TPB * 4))   // 8 float4 chunks per thread per row

#ifndef __has_builtin
#define __has_builtin(x) 0
#endif

// CDNA5 async global->LDS DMA path (ASYNCcnt-tracked), with safe fallback.
#if defined(__AMDGCN__) && __has_builtin(__builtin_amdgcn_global_load_async_to_lds_b128)
#define USE_ASYNC_LDS 1
#else
#define USE_ASYNC_LDS 0
#endif

#if USE_ASYNC_LDS
// The builtin is declared as (v4i AS1* gaddr, v4i AS3* lds, imm offset, imm cpol).
typedef int v4i __attribute__((vector_size(16)));
typedef __attribute__((address_space(1))) v4i* g_v4i_ptr;
typedef __attribute__((address_space(3))) v4i* l_v4i_ptr;
#endif

// True clang vector type so non-temporal load/store builtins accept it.
typedef float v4f __attribute__((ext_vector_type(4)));

// ---------- order-preserving f32 <-> u32 key (larger float => larger key) ----------
__device__ __forceinline__ unsigned f2key(unsigned u) {
  return (u & 0x80000000u) ? ~u : (u | 0x80000000u);
}
__device__ __forceinline__ float key2f(unsigned k) {
  unsigned u = (k & 0x80000000u) ? (k & 0x7fffffffu) : ~k;
  return __uint_as_float(u);
}

// ---------- cooperative row load: 32KB global -> LDS ----------
__device__ __forceinline__ void load_row_to_lds(const float* __restrict__ rowp,
                                                unsigned* srow) {
  const int t = threadIdx.x;
#if USE_ASYNC_LDS
  // Per-lane async DMA: each lane copies 16B global -> 16B LDS, coalesced.
#pragma unroll
  for (int c = 0; c < VEC_PER_T; ++c) {
    const int idx = (c * TPB + t) * 4;
    __builtin_amdgcn_global_load_async_to_lds_b128(
        (g_v4i_ptr)(rowp + idx),
        (l_v4i_ptr)(srow + idx),
        0, 0);
  }
#if __has_builtin(__builtin_amdgcn_s_wait_asynccnt)
  __builtin_amdgcn_s_wait_asynccnt(0);
#else
  asm volatile("s_wait_asynccnt 0" ::: "memory");
#endif
#else
  const uint4* rp = (const uint4*)rowp;
#pragma unroll
  for (int c = 0; c < VEC_PER_T; ++c) {
    const int i4 = c * TPB + t;
    uint4 v = rp[i4];
    srow[i4 * 4 + 0] = v.x;
    srow[i4 * 4 + 1] = v.y;
    srow[i4 * 4 + 2] = v.z;
    srow[i4 * 4 + 3] = v.w;
  }
#endif
  __syncthreads();
}

// ---------- exact K_TOP-th-largest of one row (block-cooperative radix select) ----
// Returns the full 32-bit key of the K_TOP-th largest element (valid in all threads).
__device__ unsigned row_select_kth(const float* __restrict__ rowp,
                                   unsigned* srow, unsigned* hist, unsigned* sres) {
  const int t = threadIdx.x;
  load_row_to_lds(rowp, srow);

  unsigned prefix = 0;
  unsigned k = K_TOP;          // rank from the top within the current prefix class
#pragma unroll
  for (int pass = 0; pass < 4; ++pass) {
    const int shift = 24 - 8 * pass;
    hist[t] = 0;               // 256 threads zero 256 bins
    __syncthreads();
    for (int i = t; i < L_ROW; i += TPB) {
      const unsigned key = f2key(srow[i]);
      const bool ok = (pass == 0) || ((key >> (shift + 8)) == prefix);
      if (ok) atomicAdd(&hist[(key >> shift) & 0xffu], 1u);
    }
    __syncthreads();
    if (t == 0) {
      // Scan bins from the top; first bin where cumulative count >= k holds
      // the k-th largest. Exact under duplicates.
      unsigned cum = 0;
      int b = 255;
      for (;;) {
        cum += hist[b];
        if (cum >= k || b == 0) break;
        --b;
      }
      sres[0] = (prefix << 8) | (unsigned)b;
      sres[1] = k - (cum - hist[b]);   // rank within selected bin
    }
    __syncthreads();
    prefix = sres[0];
    k = sres[1];
    __syncthreads();
  }
  return prefix;
}

// ---------- kernels ----------
__global__ void init_kernel(unsigned* wsM) {
  *wsM = 0xffffffffu;          // min-identity (larger than any real float key)
}

__global__ void seed_kernel(const float* __restrict__ in, unsigned* wsM) {
  __shared__ unsigned srow[L_ROW];
  __shared__ unsigned hist[TPB];
  __shared__ unsigned sres[2];
  const int row = blockIdx.x * (B_ROWS / N_SEED);   // rows 0,64,128,...
  const unsigned key = row_select_kth(in + (size_t)row * L_ROW, srow, hist, sres);
  if (threadIdx.x == 0) atomicMin(wsM, key);
}

// One block per row: cheap streaming count vs current threshold; only rows that
// could lower the global min (expected ~64 of 8192) do the exact select.
__global__ void filter_kernel(const float* __restrict__ in, unsigned* wsM) {
  __shared__ unsigned srow[L_ROW];
  __shared__ unsigned hist[TPB];
  __shared__ unsigned sres[2];
  __shared__ unsigned scount;
  const int t = threadIdx.x;
  const float* rowp = in + (size_t)blockIdx.x * L_ROW;

  // A possibly-stale (i.e. >= final) threshold is still a safe pruning bound.
  const float m = key2f(*wsM);

  if (t == 0) scount = 0;
  __syncthreads();
  unsigned cnt = 0;
  const v4f* rp4 = (const v4f*)rowp;
#pragma unroll
  for (int c = 0; c < VEC_PER_T; ++c) {
    const v4f v = rp4[c * TPB + t];
    cnt += (unsigned)(v.x >= m) + (unsigned)(v.y >= m) +
           (unsigned)(v.z >= m) + (unsigned)(v.w >= m);
  }
  atomicAdd(&scount, cnt);
  __syncthreads();

  if (scount < K_TOP) {        // row's K-th largest < m: it can move the min
    const unsigned key = row_select_kth(rowp, srow, hist, sres);
    if (t == 0) atomicMin(wsM, key);
  }
}

// Streaming pass: data is touched exactly once -> non-temporal loads and stores
// (TH=NT cache policy) so 512MB of streaming traffic does not thrash the 192MB L2.
__global__ void sigmoid_kernel(const v4f* __restrict__ in, v4f* __restrict__ out,
                               const unsigned* __restrict__ wsM, int n4) {
  const float mid = key2f(*wsM);
  int i = blockIdx.x * blockDim.x + threadIdx.x;
  const int stride = gridDim.x * blockDim.x;
  for (; i < n4; i += stride) {
    const v4f v = __builtin_nontemporal_load(in + i);
    v4f r;
    r.x = __frcp_rn(1.0f + __expf(mid - v.x));   // sigmoid(x - mid)
    r.y = __frcp_rn(1.0f + __expf(mid - v.y));
    r.z = __frcp_rn(1.0f + __expf(mid - v.z));
    r.w = __frcp_rn(1.0f + __expf(mid - v.w));
    __builtin_nontemporal_store(r, out + i);
  }
}

extern "C" void kernel_launch(void* const* d_in, const int* in_sizes, int n_in,
                              void* d_out, int out_size, void* d_ws, size_t ws_size,
                              hipStream_t stream) {
  (void)in_sizes; (void)n_in; (void)out_size; (void)ws_size;
  const float* in = (const float*)d_in[0];
  float* out = (float*)d_out;
  unsigned* wsM = (unsigned*)d_ws;   // 4 bytes: running min key

  init_kernel<<<1, 1, 0, stream>>>(wsM);
  seed_kernel<<<N_SEED, TPB, 0, stream>>>(in, wsM);
  filter_kernel<<<B_ROWS, TPB, 0, stream>>>(in, wsM);
  const int n4 = (B_ROWS * L_ROW) / 4;
  sigmoid_kernel<<<8192, TPB, 0, stream>>>((const v4f*)in, (v4f*)out, wsM, n4);
}